// GATModel_55946243998130
// MI455X (gfx1250) — compile-verified
//
#include <hip/hip_runtime.h>
#include <cstdint>

typedef __attribute__((ext_vector_type(2))) float v2f;
typedef __attribute__((ext_vector_type(8))) float v8f;

#define NEG_SLOPE 0.2f

// ---------------------------------------------------------------------------
// WMMA GEMM: C[nrows x NC] = A[nrows x K] * B[K x NC]  (all row-major, f32)
// One wave computes a 16-row x NC tile using V_WMMA_F32_16X16X4_F32.
// B (<= 32KB) is staged in LDS once per block. (Codegen verified: pipelined
// v_wmma_f32_16x16x4_f32 against ds_load_2addr_b32.)
// ---------------------------------------------------------------------------
template <int K, int NC>
__global__ __launch_bounds__(256) void gat_gemm_wmma(
    const float* __restrict__ A, const float* __restrict__ B,
    float* __restrict__ C, int nrows) {
  __shared__ float Bs[K * NC];
  for (int i = threadIdx.x; i < K * NC; i += 256) Bs[i] = B[i];
  __syncthreads();

  const int lane = threadIdx.x & 31;
  const int wave = threadIdx.x >> 5;
  const int tile = blockIdx.x * 8 + wave;  // 16 rows per tile, 8 waves/block
  if (tile * 16 >= nrows) return;          // wave-uniform: EXEC stays all-ones

  const int m  = lane & 15;        // A row within tile; B/C column within tile
  const int kh = (lane >> 4) * 2;  // K sub-offset: 0 for lanes 0-15, 2 for 16-31

  const float* Arow = A + (size_t)(tile * 16 + m) * K;

  v8f zero = {};
  v8f acc[NC / 16];
#pragma unroll
  for (int t = 0; t < NC / 16; ++t) acc[t] = zero;

#pragma unroll 4
  for (int k0 = 0; k0 < K; k0 += 4) {
    v2f a;
    a.x = Arow[k0 + kh + 0];
    a.y = Arow[k0 + kh + 1];
#pragma unroll
    for (int t = 0; t < NC / 16; ++t) {
      v2f b;
      b.x = Bs[(k0 + kh + 0) * NC + t * 16 + m];
      b.y = Bs[(k0 + kh + 1) * NC + t * 16 + m];
      acc[t] = __builtin_amdgcn_wmma_f32_16x16x4_f32(
          /*neg_a=*/false, a, /*neg_b=*/false, b,
          /*c_mod=*/(short)0, acc[t], /*reuse_a=*/false, /*reuse_b=*/false);
    }
  }

  // C/D layout: VGPR r, lanes 0-15 -> row r; lanes 16-31 -> row r+8
  const int rbase = tile * 16 + ((lane >> 4) << 3);
#pragma unroll
  for (int t = 0; t < NC / 16; ++t) {
#pragma unroll
    for (int r = 0; r < 8; ++r) {
      C[(size_t)(rbase + r) * NC + t * 16 + m] = acc[t][r];
    }
  }
}

// ---------------------------------------------------------------------------
// Edge decode with appended self-loops.
// ---------------------------------------------------------------------------
__device__ __forceinline__ void edge_nodes(const long long* __restrict__ ei,
                                           long long E, long long e,
                                           int& sn, int& dn) {
  if (e < E) {
    sn = (int)ei[e];
    dn = (int)ei[E + e];
  } else {
    sn = dn = (int)(e - E);
  }
}

// ---------------------------------------------------------------------------
// CSR build: degree histogram -> exclusive scan -> fill (stores src node id).
// ---------------------------------------------------------------------------
__global__ void gat_zero_u32(unsigned* __restrict__ p, long long n) {
  long long tid = (long long)blockIdx.x * blockDim.x + threadIdx.x;
  if (tid < n) p[tid] = 0u;
}

__global__ void gat_degree(const long long* __restrict__ ei, long long E,
                           long long Etot, unsigned* __restrict__ deg) {
  long long tid = (long long)blockIdx.x * blockDim.x + threadIdx.x;
  if (tid >= Etot) return;
  int sn, dn;
  edge_nodes(ei, E, tid, sn, dn);
  atomicAdd(&deg[dn], 1u);
}

// Single-workgroup chunked Hillis-Steele exclusive scan.
// Reads deg from `deg_cursor`, writes exclusive prefix to both `indptr`
// and back into `deg_cursor` (which then serves as the fill cursor).
__global__ __launch_bounds__(1024) void gat_scan(
    unsigned* __restrict__ deg_cursor, unsigned* __restrict__ indptr, int N) {
  __shared__ unsigned buf[1024];
  __shared__ unsigned carry;
  if (threadIdx.x == 0) carry = 0u;
  __syncthreads();
  for (int base = 0; base < N; base += 1024) {
    int i = base + (int)threadIdx.x;
    unsigned v = (i < N) ? deg_cursor[i] : 0u;
    buf[threadIdx.x] = v;
    __syncthreads();
    for (int off = 1; off < 1024; off <<= 1) {
      unsigned t = (threadIdx.x >= (unsigned)off) ? buf[threadIdx.x - off] : 0u;
      __syncthreads();
      buf[threadIdx.x] += t;
      __syncthreads();
    }
    unsigned excl = buf[threadIdx.x] - v + carry;
    if (i < N) {
      indptr[i] = excl;
      deg_cursor[i] = excl;
    }
    __syncthreads();
    if (threadIdx.x == 0) carry += buf[1023];
    __syncthreads();
  }
  if (threadIdx.x == 0) indptr[N] = carry;
}

__global__ void gat_fill(const long long* __restrict__ ei, long long E,
                         long long Etot, unsigned* __restrict__ cursor,
                         int* __restrict__ esrc) {
  long long tid = (long long)blockIdx.x * blockDim.x + threadIdx.x;
  if (tid >= Etot) return;
  int sn, dn;
  edge_nodes(ei, E, tid, sn, dn);
  unsigned pos = atomicAdd(&cursor[dn], 1u);
  esrc[pos] = sn;
}

// ---------------------------------------------------------------------------
// Attention logits per node: alpha_src/dst[n,h] = sum_f h[n,h,f] * a[h,f]
// ---------------------------------------------------------------------------
__global__ void gat_alpha(const float* __restrict__ h,
                          const float* __restrict__ a_src,
                          const float* __restrict__ a_dst,
                          float* __restrict__ as, float* __restrict__ ad,
                          int N, int H, int F) {
  int tid = blockIdx.x * blockDim.x + threadIdx.x;
  if (tid >= N * H) return;
  int n = tid / H, hh = tid % H;
  const float* hp  = h + ((size_t)n * H + hh) * F;
  const float* asp = a_src + (size_t)hh * F;
  const float* adp = a_dst + (size_t)hh * F;
  float d1 = 0.f, d2 = 0.f;
  for (int f = 0; f < F; ++f) {
    float v = hp[f];
    d1 += v * asp[f];
    d2 += v * adp[f];
  }
  as[tid] = d1;
  ad[tid] = d2;
}

// ---------------------------------------------------------------------------
// Per-(dst,head) segment softmax over the CSR slice, no atomics.
// Pass 1: e = leaky_relu(as[src]+ad[dst]) -> ev, track max.
// Pass 2: ev = exp(e - max), accumulate sum.
// Pass 3: ev *= 1/sum  => ev holds final attention weights alpha.
// (deg >= 1 always, thanks to self-loops.)
// ---------------------------------------------------------------------------
__global__ void gat_softmax_csr(const unsigned* __restrict__ indptr,
                                const int* __restrict__ esrc,
                                const float* __restrict__ as,
                                const float* __restrict__ ad,
                                float* __restrict__ ev, int N, int H) {
  int tid = blockIdx.x * blockDim.x + threadIdx.x;
  if (tid >= N * H) return;
  int n = tid / H, hh = tid % H;
  const unsigned p0 = indptr[n], p1 = indptr[n + 1];
  const float adv = ad[(size_t)n * H + hh];
  float mx = -__builtin_inff();
  for (unsigned p = p0; p < p1; ++p) {
    float x = as[(size_t)esrc[p] * H + hh] + adv;
    x = (x > 0.f) ? x : NEG_SLOPE * x;
    ev[(size_t)p * H + hh] = x;
    mx = fmaxf(mx, x);
  }
  float sum = 0.f;
  for (unsigned p = p0; p < p1; ++p) {
    float ex = expf(ev[(size_t)p * H + hh] - mx);
    ev[(size_t)p * H + hh] = ex;
    sum += ex;
  }
  const float inv = 1.f / sum;
  for (unsigned p = p0; p < p1; ++p) ev[(size_t)p * H + hh] *= inv;
}

// ---------------------------------------------------------------------------
// Destination-grouped aggregation + fused bias (+ReLU): one wave32 per dst.
// Each lane owns FT/32 contiguous features -> coalesced b64/b128 gathers of
// h[src] (table resident in the 192MB L2). Each output written exactly once.
// ---------------------------------------------------------------------------
template <int FT, int H>
__global__ __launch_bounds__(256) void gat_agg_csr(
    const unsigned* __restrict__ indptr, const int* __restrict__ esrc,
    const float* __restrict__ h, const float* __restrict__ ev,
    const float* __restrict__ bias, float* __restrict__ out, int N, int relu) {
  constexpr int PER = FT / 32;
  constexpr int F = FT / H;
  const int lane = threadIdx.x & 31;
  const int wave = threadIdx.x >> 5;
  const int n = blockIdx.x * 8 + wave;
  if (n >= N) return;
  const int f0 = lane * PER;

  float acc[PER];
#pragma unroll
  for (int j = 0; j < PER; ++j) acc[j] = 0.f;

  const unsigned p0 = indptr[n], p1 = indptr[n + 1];
  for (unsigned p = p0; p < p1; ++p) {
    const int sn = esrc[p];
    const float* hp = h + (size_t)sn * FT + f0;
#pragma unroll
    for (int j = 0; j < PER; ++j) {
      const float alpha = ev[(size_t)p * H + (f0 + j) / F];
      acc[j] += hp[j] * alpha;
    }
  }

#pragma unroll
  for (int j = 0; j < PER; ++j) {
    float v = acc[j] + bias[f0 + j];
    if (relu) v = v > 0.f ? v : 0.f;
    out[(size_t)n * FT + f0 + j] = v;
  }
}

// ---------------------------------------------------------------------------
extern "C" void kernel_launch(void* const* d_in, const int* in_sizes, int n_in,
                              void* d_out, int out_size, void* d_ws,
                              size_t ws_size, hipStream_t stream) {
  const float* x      = (const float*)d_in[0];
  const long long* ei = (const long long*)d_in[1];  // int64 [2, E]
  const float* W1     = (const float*)d_in[2];      // [128, 64]
  const float* att_s1 = (const float*)d_in[3];      // [2, 32]
  const float* att_d1 = (const float*)d_in[4];
  const float* b1     = (const float*)d_in[5];      // [64]
  const float* W2     = (const float*)d_in[6];      // [64, 128]
  const float* att_s2 = (const float*)d_in[7];      // [1, 128]
  const float* att_d2 = (const float*)d_in[8];
  const float* b2     = (const float*)d_in[9];      // [128]

  const int N = in_sizes[0] / 128;
  const long long E = in_sizes[1] / 2;
  const long long Etot = E + N;

  // ---- Workspace layout ----
  // Integer CSR region (persistent across both layers):
  unsigned* indptr = (unsigned*)d_ws;            // N+1
  unsigned* cursor = indptr + (N + 1);           // N (degree, then cursor)
  int* esrc        = (int*)(cursor + N);         // Etot
  // Float region:
  float* fb    = (float*)(esrc + Etot);
  float* hrelu = fb;                             // N*64, live across phases
  float* pA    = hrelu + (size_t)N * 64;
  // phase A
  float* h1  = pA;                               // N*64
  float* as1 = h1 + (size_t)N * 64;              // N*2
  float* ad1 = as1 + (size_t)N * 2;              // N*2
  float* ev1 = ad1 + (size_t)N * 2;              // Etot*2
  // phase B (overlaps dead phase-A temps)
  float* h2  = pA;                               // N*128
  float* as2 = h2 + (size_t)N * 128;             // N
  float* ad2 = as2 + (size_t)N;                  // N
  float* ev2 = ad2 + (size_t)N;                  // Etot

  const int TB = 256;
  auto cdiv = [](long long a, long long b) { return (int)((a + b - 1) / b); };

  // ---- Build CSR once (shared by both layers) ----
  gat_zero_u32<<<cdiv(N, TB), TB, 0, stream>>>(cursor, N);
  gat_degree<<<cdiv(Etot, TB), TB, 0, stream>>>(ei, E, Etot, cursor);
  gat_scan<<<1, 1024, 0, stream>>>(cursor, indptr, N);
  gat_fill<<<cdiv(Etot, TB), TB, 0, stream>>>(ei, E, Etot, cursor, esrc);

  // ---- Layer 1 (H=2, F=32, feature width 64) ----
  gat_gemm_wmma<128, 64><<<cdiv(N, 128), 256, 0, stream>>>(x, W1, h1, N);
  gat_alpha<<<cdiv((long long)N * 2, TB), TB, 0, stream>>>(h1, att_s1, att_d1,
                                                           as1, ad1, N, 2, 32);
  gat_softmax_csr<<<cdiv((long long)N * 2, TB), TB, 0, stream>>>(
      indptr, esrc, as1, ad1, ev1, N, 2);
  gat_agg_csr<64, 2><<<cdiv(N, 8), 256, 0, stream>>>(indptr, esrc, h1, ev1, b1,
                                                     hrelu, N, /*relu=*/1);

  // ---- Layer 2 (H=1, F=128) ----
  gat_gemm_wmma<64, 128><<<cdiv(N, 128), 256, 0, stream>>>(hrelu, W2, h2, N);
  gat_alpha<<<cdiv((long long)N, TB), TB, 0, stream>>>(h2, att_s2, att_d2, as2,
                                                       ad2, N, 1, 128);
  gat_softmax_csr<<<cdiv((long long)N, TB), TB, 0, stream>>>(indptr, esrc, as2,
                                                             ad2, ev2, N, 1);
  gat_agg_csr<128, 1><<<cdiv(N, 8), 256, 0, stream>>>(
      indptr, esrc, h2, ev2, b2, (float*)d_out, N, /*relu=*/0);
}